// GAGQueryAndGroup_48215302865114
// MI455X (gfx1250) — compile-verified
//
#include <hip/hip_runtime.h>
#include <hip/hip_bf16.h>
#include <stdint.h>

// Problem constants (fixed by the reference)
#define BQ    4
#define NPTS  16384
#define NQ    1024
#define NCH   64
#define NS    32
// radius^2 = 0.04, (lambda*radius)^2 = 0.01

typedef __attribute__((ext_vector_type(2))) float v2f;
typedef __attribute__((ext_vector_type(8))) float v8f;

static __device__ __forceinline__ unsigned ballot32(bool p) {
#if __has_builtin(__builtin_amdgcn_ballot_w32)
    return __builtin_amdgcn_ballot_w32(p);
#else
    return (unsigned)__ballot(p);
#endif
}

// Select x[lane & 7] from 8 wave-uniform values with a 3-level cndmask tree
// (constant indices after inlining -> no scratch, no LDS, no barriers).
static __device__ __forceinline__ unsigned sel8(const unsigned x[8], int lane) {
    const unsigned t0 = (lane & 1) ? x[1] : x[0];
    const unsigned t1 = (lane & 1) ? x[3] : x[2];
    const unsigned t2 = (lane & 1) ? x[5] : x[4];
    const unsigned t3 = (lane & 1) ? x[7] : x[6];
    const unsigned u0 = (lane & 2) ? t1 : t0;
    const unsigned u1 = (lane & 2) ? t3 : t2;
    return (lane & 4) ? u1 : u0;
}

// ---------------------------------------------------------------------------
// Kernel 1: geometry-aware ball query.
// One wave32 handles 16 query points. Each iteration evaluates TWO 16x16
// tiles of squared distances (32 points) with V_WMMA_F32_16X16X4_F32:
//   D[m][n] = [qx qy qz 1] . [-2px -2py -2pz |p|^2]^T + |q_m|^2 = |q_m-p_n|^2
// VGPR layouts (ISA 7.12.2):
//   A 16x4 : v0 = K0 (lanes 0-15) / K2 (lanes 16-31); v1 = K1 / K3
//   B 4x16 : v0 = K0 row (lanes 0-15) / K2 row (lanes 16-31); v1 = K1 / K3
//   C/D    : vgpr v, lane l -> row m = v + 8*(l>=16), col n = l&15
// Ballot of (d2 < r2) over D-vgpr v yields the 16-bit hit masks of query
// rows v (low half) and v+8 (high half); lane m grabs its row's mask with a
// uniform select tree, concatenates both tiles (increasing point order), and
// does the ordered first-NS extraction, with a wave-wide early exit.
// ---------------------------------------------------------------------------
__global__ __launch_bounds__(32)
void gag_query_kernel(const float* __restrict__ xyz,
                      const float* __restrict__ new_xyz,
                      const int*   __restrict__ comp,
                      const int*   __restrict__ ncomp,
                      int*         __restrict__ idx_out)
{
    __shared__ int      s_idx[16][NS];
    __shared__ float    s_qn[16];
    __shared__ int      s_qc[16];

    const int lane  = threadIdx.x;      // 0..31
    const int hi    = lane >> 4;        // upper half-wave?
    const int lq    = lane & 15;
    const int tile  = blockIdx.x;       // B * (NQ/16) tiles
    const int b     = tile / (NQ / 16);
    const int qbase = (tile % (NQ / 16)) * 16;

    // Load this half-wave's query point (lq-th query of the tile).
    const float* qp = new_xyz + ((size_t)b * NQ + qbase + lq) * 3;
    const float qx = qp[0], qy = qp[1], qz = qp[2];
    if (lane < 16) {
        s_qn[lane] = qx * qx + qy * qy + qz * qz;
        s_qc[lane] = ncomp[b * NQ + qbase + lane];
    }
    __syncthreads();

    // A matrix (constant across the whole scan).
    v2f amat;
    amat[0] = hi ? qz : qx;   // K2 : K0
    amat[1] = hi ? 1.0f : qy; // K3 : K1

    // Per-lane copies of the row-indexed C values / query components.
    float cvals[8];
    int   qcs[8];
#pragma unroll
    for (int v = 0; v < 8; ++v) {
        cvals[v] = s_qn[v + 8 * hi];
        qcs[v]   = s_qc[v + 8 * hi];
    }

    // Lanes 0-15 own query rows; lanes 16-31 start "done" for the exit vote.
    int cnt = (lane < 16) ? 0 : NS;
    const int shamt = (lane & 8) ? 16 : 0;   // row >= 8 -> high half of ballot

    for (int base = 0; base < NPTS; base += 32) {
        // Tile 0: points base..base+15, tile 1: base+16..base+31.
        // Both half-waves read the same 16 points per tile (L0/L2 resident)
        // and repack into B layout.
        const float* pp0 = xyz + ((size_t)b * NPTS + base + lq) * 3;
        const float p0x = pp0[0], p0y = pp0[1], p0z = pp0[2];
        const float p1x = pp0[48], p1y = pp0[49], p1z = pp0[50]; // +16 pts
        const int   pc0 = comp[b * NPTS + base + lq];
        const int   pc1 = comp[b * NPTS + base + 16 + lq];

        v2f b0, b1;
        b0[0] = hi ? (-2.0f * p0z) : (-2.0f * p0x);
        b0[1] = hi ? (p0x * p0x + p0y * p0y + p0z * p0z) : (-2.0f * p0y);
        b1[0] = hi ? (-2.0f * p1z) : (-2.0f * p1x);
        b1[1] = hi ? (p1x * p1x + p1y * p1y + p1z * p1z) : (-2.0f * p1y);

        v8f cmat;
#pragma unroll
        for (int v = 0; v < 8; ++v) cmat[v] = cvals[v];

        const v8f d0 = __builtin_amdgcn_wmma_f32_16x16x4_f32(
            false, amat, false, b0, (short)0, cmat, false, false);
        const v8f d1 = __builtin_amdgcn_wmma_f32_16x16x4_f32(
            false, amat, false, b1, (short)0, cmat, false, false);

        // Validity ballots; column n = lane&15 -> this lane's point component
        // is exactly the one needed for the per-element radius choice.
        unsigned bal0[8], bal1[8];
#pragma unroll
        for (int v = 0; v < 8; ++v) {
            bal0[v] = ballot32(d0[v] < ((qcs[v] == pc0) ? 0.04f : 0.01f));
            bal1[v] = ballot32(d1[v] < ((qcs[v] == pc1) ? 0.04f : 0.01f));
        }

        // Lane m extracts its row's 16-bit mask from each tile and scans the
        // combined 32-bit mask in increasing point order.
        const unsigned r0 = (sel8(bal0, lane) >> shamt) & 0xFFFFu;
        const unsigned r1 = (sel8(bal1, lane) >> shamt) & 0xFFFFu;
        unsigned m32 = r0 | (r1 << 16);
        if (lane < 16 && cnt < NS) {
            while (m32 && cnt < NS) {
                s_idx[lane][cnt++] = base + __builtin_ctz(m32);
                m32 &= m32 - 1u;
            }
        }

        // Early exit once all 16 queries have NS samples.
        if (ballot32(cnt >= NS) == 0xFFFFFFFFu) break;
    }

    // Pad with first valid index (0 if the ball is empty) — pointnet2 style.
    if (lane < 16) {
        const int first = (cnt > 0) ? s_idx[lane][0] : 0;
        for (int s = cnt; s < NS; ++s) s_idx[lane][s] = first;
    }
    __syncthreads();

    // 16*NS = 512 ints, coalesced.
    int* dst = idx_out + ((size_t)b * NQ + qbase) * NS;
    for (int i = lane; i < 16 * NS; i += 32)
        dst[i] = s_idx[i >> 5][i & 31];
}

// ---------------------------------------------------------------------------
// Kernel 2: grouping gather. One thread per output element; the nsample axis
// maps to consecutive lanes so the 35 MB store is fully coalesced; gathers
// hit the L2-resident inputs.
// out[b][ch][j][s], ch<3 -> recentered xyz, ch>=3 -> features[ch-3].
// ---------------------------------------------------------------------------
__global__ __launch_bounds__(256)
void gag_group_kernel(const float* __restrict__ xyz,
                      const float* __restrict__ new_xyz,
                      const float* __restrict__ feat,
                      const int*   __restrict__ idx,
                      float*       __restrict__ out)
{
    const int64_t gid   = (int64_t)blockIdx.x * 256 + threadIdx.x;
    const int64_t total = (int64_t)BQ * (3 + NCH) * NQ * NS;
    if (gid >= total) return;

    const int     s   = (int)(gid & (NS - 1));
    const int64_t row = gid >> 5;
    const int     j   = (int)(row % NQ);
    const int64_t t   = row / NQ;
    const int     ch  = (int)(t % (3 + NCH));
    const int     b   = (int)(t / (3 + NCH));

    const int k = idx[((int64_t)b * NQ + j) * NS + s];

    float val;
    if (ch < 3) {
        val = xyz[((int64_t)b * NPTS + k) * 3 + ch] -
              new_xyz[((int64_t)b * NQ + j) * 3 + ch];
    } else {
        val = feat[((int64_t)b * NCH + (ch - 3)) * (int64_t)NPTS + k];
    }
    out[gid] = val;
}

// ---------------------------------------------------------------------------
extern "C" void kernel_launch(void* const* d_in, const int* in_sizes, int n_in,
                              void* d_out, int out_size, void* d_ws, size_t ws_size,
                              hipStream_t stream)
{
    (void)in_sizes; (void)n_in; (void)out_size; (void)ws_size;

    const float* xyz     = (const float*)d_in[0];
    const float* new_xyz = (const float*)d_in[1];
    const int*   comp    = (const int*)d_in[2];
    const int*   ncomp   = (const int*)d_in[3];
    const float* feat    = (const float*)d_in[4];
    float*       out     = (float*)d_out;

    // Workspace: B*NQ*NS int32 indices = 512 KB.
    int* idx_ws = (int*)d_ws;

    gag_query_kernel<<<BQ * (NQ / 16), 32, 0, stream>>>(
        xyz, new_xyz, comp, ncomp, idx_ws);

    const int64_t total  = (int64_t)BQ * (3 + NCH) * NQ * NS;
    const int     blocks = (int)((total + 255) / 256);
    gag_group_kernel<<<blocks, 256, 0, stream>>>(
        xyz, new_xyz, feat, idx_ws, out);
}